// ExpModel_77824807403811
// MI455X (gfx1250) — compile-verified
//
#include <hip/hip_runtime.h>

typedef __attribute__((ext_vector_type(2))) float v2f;
typedef __attribute__((ext_vector_type(8))) float v8f;
typedef __attribute__((ext_vector_type(4))) unsigned int v4u;
typedef __attribute__((ext_vector_type(8))) int v8i;
typedef __attribute__((ext_vector_type(4))) int v4i;

#define NQ        25
#define DEPTH     8
#define NSTATE    (1u << NQ)
#define TILE_BITS 15
#define TILE      (1u << TILE_BITS)     // 32768 floats = 128 KB per tile
#define THREADS   256                   // 8 wave32
#define WAVES     (THREADS / 32)
#define NMATS     (DEPTH * 7)

#if __has_builtin(__builtin_amdgcn_tensor_load_to_lds)
#define HAVE_TDM 1
#else
#define HAVE_TDM 0
#endif

// ---- CDNA5 async global<->LDS helpers (ASYNCcnt) ----
__device__ __forceinline__ void async_load_b128(unsigned lds_byte_off, const void* gptr) {
    asm volatile("global_load_async_to_lds_b128 %0, %1, off"
                 :: "v"(lds_byte_off),
                    "v"((unsigned long long)(uintptr_t)gptr)
                 : "memory");
}
__device__ __forceinline__ void async_store_b128(void* gptr, unsigned lds_byte_off) {
    asm volatile("global_store_async_from_lds_b128 %0, %1, off"
                 :: "v"((unsigned long long)(uintptr_t)gptr),
                    "v"(lds_byte_off)
                 : "memory");
}
__device__ __forceinline__ void wait_async0() {
    asm volatile("s_wait_asynccnt 0x0" ::: "memory");
}

// Fused 16x16 gate matrices: for group g, the 4 sub-bit slots (k=0 lowest)
// hold qubit index (theta column) or -1 for identity padding.
__device__ __constant__ int g_qubits[7][4] = {
    { 3,  2,  1,  0},   // bits 21,22,23,24
    { 7,  6,  5,  4},   // bits 17,18,19,20
    {10,  9,  8, -1},   // bits 14,15,16,(17=I)
    {14, 13, 12, 11},   // bits 10,11,12,13
    {18, 17, 16, 15},   // bits 6,7,8,9
    {22, 21, 20, 19},   // bits 2,3,4,5
    {24, 23, -1, -1},   // bits 0,1,(2,3=I)
};

// U[r][c] = prod_k M_k[r_k][c_k], M = [[cos,-sin],[sin,cos]](theta/2) or I.
__global__ void build_mats_kernel(const float* __restrict__ theta,
                                  float* __restrict__ mats) {
    int m = blockIdx.x;                 // layer*7 + g
    int layer = m / 7, g = m % 7;
    int tid = threadIdx.x;              // r*16 + c
    int r = tid >> 4, c = tid & 15;
    float v = 1.0f;
#pragma unroll
    for (int k = 0; k < 4; ++k) {
        int q = g_qubits[g][k];
        int rb = (r >> k) & 1, cb = (c >> k) & 1;
        float f;
        if (q < 0) {
            f = (rb == cb) ? 1.0f : 0.0f;
        } else {
            float a  = 0.5f * theta[layer * NQ + q];
            float cs = __cosf(a), sn = __sinf(a);
            f = (rb == 0) ? ((cb == 0) ? cs : -sn)
                          : ((cb == 0) ? sn :  cs);
        }
        v *= f;
    }
    mats[m * 256 + tid] = v;
}

__global__ void init_state_kernel(float4* __restrict__ state4,
                                  float* __restrict__ out) {
    unsigned i = blockIdx.x * blockDim.x + threadIdx.x;
    float4 z = make_float4(0.f, 0.f, 0.f, 0.f);
    if (i == 0) { z.x = 1.0f; out[0] = 0.0f; }
    state4[i] = z;
}

// Apply one fused 16x16 matrix U to the LDS tile on local bits [b..b+3]
// via V_WMMA_F32_16X16X4_F32 (4 K-chunks). Column-tiles are partitioned
// across waves with a wave-uniform (scalar) loop so EXEC stays all-1s.
__device__ __forceinline__ void apply_group(float* __restrict__ lds,
                                            const float* __restrict__ U,
                                            int b) {
    const int lane  = threadIdx.x & 31;
    const int wave  = __builtin_amdgcn_readfirstlane((int)(threadIdx.x >> 5));
    const int n     = lane & 15;     // A row m / B-D column n
    const int khalf = lane >> 4;     // 0: K=k0,k0+1  1: K=k0+2,k0+3

    v2f a[4];
#pragma unroll
    for (int kk = 0; kk < 4; ++kk) {
        int k = kk * 4 + khalf * 2;
        a[kk].x = U[n * 16 + k];
        a[kk].y = U[n * 16 + k + 1];
    }

    const int ntiles  = (TILE >> 4) >> 4;   // 128 column-tiles of 16
    const int lowmask = (1 << b) - 1;
#pragma unroll 1
    for (int ct = wave; ct < ntiles; ct += WAVES) {
        int c  = ct * 16 + n;
        int Lb = ((c >> b) << (b + 4)) | (c & lowmask);  // row slot cleared
        v8f d = {0.f, 0.f, 0.f, 0.f, 0.f, 0.f, 0.f, 0.f};
#pragma unroll
        for (int kk = 0; kk < 4; ++kk) {
            int k = kk * 4 + khalf * 2;
            v2f bb;
            bb.x = lds[Lb + (k << b)];
            bb.y = lds[Lb + ((k + 1) << b)];
            d = __builtin_amdgcn_wmma_f32_16x16x4_f32(
                    false, a[kk], false, bb, (short)0, d, false, false);
        }
#pragma unroll
        for (int j = 0; j < 8; ++j) {
            int row = j + khalf * 8;
            lds[Lb + (row << b)] = d[j];
        }
    }
}

// PHASE 0: tile = all bits 14..24 x 16 contiguous floats; blockIdx = bits 4..13.
//   Tile fill is a 2D strided gather (2048 rows x 64B, stride 64KB) -> TDM.
//   RY q0..q10 then CNOT 0..9 (suffix-xor of hi bits folded into async store).
// PHASE 1: tile = contiguous 2^15 floats; blockIdx = bits 15..24.
//   RY q11..q24 then CNOT 10..23 (15-bit Gray permutation folded into store).
// REDUCE: last pass skips the store and accumulates sign * |amp|^2.
template <int PHASE, int REDUCE>
__global__ void apply_layer_kernel(float* __restrict__ state,
                                   const float* __restrict__ mats,
                                   int layer,
                                   float* __restrict__ out) {
    extern __shared__ float lds[];          // TILE + 1024 floats
    float*  ldsU   = lds + TILE;
    float4* lds4   = reinterpret_cast<float4*>(lds);
    float4* state4 = reinterpret_cast<float4*>(state);
    const int tid = threadIdx.x;
    const unsigned blk = blockIdx.x;
    const unsigned ldsbase = (unsigned)(uintptr_t)lds;   // LDS byte offset of tile

    {   // stage this phase's fused matrices into LDS
        const int g0 = (PHASE == 0) ? 0 : 3;
        const int ng = (PHASE == 0) ? 3 : 4;
        const float* src = mats + (layer * 7 + g0) * 256;
        for (int i = tid; i < ng * 256; i += THREADS) ldsU[i] = src[i];
    }

    if (PHASE == 0) {
#if HAVE_TDM
        if (tid < 32) {   // one TDM descriptor per workgroup, issued by wave 0
            unsigned long long ga =
                (unsigned long long)(uintptr_t)(state + ((size_t)blk << 4));
            v4u g0 = { 1u,                                   // count=1, user D#
                       ldsbase,                              // lds_addr (bytes)
                       (unsigned)(ga & 0xffffffffu),
                       (unsigned)((ga >> 32) & 0x01ffffffu) | (2u << 30) }; // type=2
            v8i g1 = { (int)(2u << 16),       // data_size = 4B, no multicast
                       (int)(16u << 16),      // tensor_dim0 = 16 elems
                       (int)(2048u << 16),    // tensor_dim1 = 2048 rows
                       (int)(16u << 16),      // tile_dim0 = 16 elems (64B row)
                       (int)2048,             // tile_dim1 = 2048 rows
                       (int)(1u << 14),       // tensor_dim0_stride = 16384 elems
                       0, 0 };
            v4i gz = { 0, 0, 0, 0 };
#if defined(__clang_major__) && (__clang_major__ >= 23)
            v8i gz8 = { 0, 0, 0, 0, 0, 0, 0, 0 };
            __builtin_amdgcn_tensor_load_to_lds(g0, g1, gz, gz, gz8, 0);
#else
            __builtin_amdgcn_tensor_load_to_lds(g0, g1, gz, gz, 0);
#endif
            __builtin_amdgcn_s_wait_tensorcnt(0);
        }
#else
        for (int f = tid; f < (int)(TILE / 4); f += THREADS) {
            unsigned hi = (unsigned)f >> 2, q = (unsigned)f & 3;
            async_load_b128(ldsbase + 16u * (unsigned)f,
                            (const void*)(state4 + (((size_t)hi << 12) | (blk << 2) | q)));
        }
        wait_async0();
#endif
    } else {
        for (int f = tid; f < (int)(TILE / 4); f += THREADS)
            async_load_b128(ldsbase + 16u * (unsigned)f,
                            (const void*)(state4 + (((size_t)blk << 13) + f)));
        wait_async0();
    }
    __syncthreads();

    if (PHASE == 0) {
        apply_group(lds, ldsU + 0 * 256, 11);  __syncthreads();  // bits 21..24
        apply_group(lds, ldsU + 1 * 256,  7);  __syncthreads();  // bits 17..20
        apply_group(lds, ldsU + 2 * 256,  4);  __syncthreads();  // bits 14..17
    } else {
        apply_group(lds, ldsU + 0 * 256, 10);  __syncthreads();  // bits 10..13
        apply_group(lds, ldsU + 1 * 256,  6);  __syncthreads();  // bits 6..9
        apply_group(lds, ldsU + 2 * 256,  2);  __syncthreads();  // bits 2..5
        apply_group(lds, ldsU + 3 * 256,  0);  __syncthreads();  // bits 0..3
    }

    if (REDUCE) {
        // |amp|^2 is invariant under the intra-tile CNOT permutation and
        // bit24 (measured qubit) is never a CNOT target -> sign is per-block.
        float acc = 0.f;
        for (int f = tid; f < (int)(TILE / 4); f += THREADS) {
            float4 v = lds4[f];
            acc += v.x * v.x + v.y * v.y + v.z * v.z + v.w * v.w;
        }
        acc *= ((blk >> 9) & 1u) ? -1.0f : 1.0f;
#pragma unroll
        for (int off = 16; off > 0; off >>= 1)
            acc += __shfl_down(acc, off, 32);
        if ((tid & 31) == 0) atomicAdd(out, acc);
    } else if (PHASE == 0) {
        // composite CNOT 0..9: new[t] = old[t ^ (t>>1)] on the 11 hi bits;
        // source quads are contiguous in LDS -> stream LDS->HBM directly.
        for (int f = tid; f < (int)(TILE / 4); f += THREADS) {
            unsigned t = (unsigned)f >> 2, q = (unsigned)f & 3;
            unsigned src4 = ((t ^ (t >> 1)) << 2) | q;
            async_store_b128((void*)(state4 + (((size_t)t << 12) | (blk << 2) | q)),
                             ldsbase + 16u * src4);
        }
        wait_async0();
    } else {
        // composite CNOT 10..23: 15-bit Gray gather; the 4 sources of one
        // output float4 live in one aligned LDS quad but permuted within it,
        // so swizzle through VGPRs.
        for (int f = tid; f < (int)(TILE / 4); f += THREADS) {
            unsigned t = (unsigned)f << 2;
            unsigned g = t ^ (t >> 1);
            float4 v;
            v.x = lds[g];
            v.y = lds[g ^ 1u];
            v.z = lds[g ^ 3u];
            v.w = lds[g ^ 2u];
            state4[((size_t)blk << 13) + f] = v;
        }
    }
}

extern "C" void kernel_launch(void* const* d_in, const int* in_sizes, int n_in,
                              void* d_out, int out_size, void* d_ws, size_t ws_size,
                              hipStream_t stream) {
    (void)in_sizes; (void)n_in; (void)out_size; (void)ws_size;
    const float* theta = (const float*)d_in[0];
    float* out   = (float*)d_out;
    float* mats  = (float*)d_ws;                    // 56*256 f32 = 56 KB
    float* state = (float*)d_ws + (1 << 16);        // 256 KB offset, 128 MB state

    const size_t ldsb = (TILE + 1024) * sizeof(float);  // 132 KB -> 2 blocks/WGP
    hipFuncSetAttribute((const void*)&apply_layer_kernel<0, 0>,
                        hipFuncAttributeMaxDynamicSharedMemorySize, (int)ldsb);
    hipFuncSetAttribute((const void*)&apply_layer_kernel<1, 0>,
                        hipFuncAttributeMaxDynamicSharedMemorySize, (int)ldsb);
    hipFuncSetAttribute((const void*)&apply_layer_kernel<1, 1>,
                        hipFuncAttributeMaxDynamicSharedMemorySize, (int)ldsb);

    build_mats_kernel<<<NMATS, 256, 0, stream>>>(theta, mats);
    init_state_kernel<<<(NSTATE / 4) / 256, 256, 0, stream>>>(
        reinterpret_cast<float4*>(state), out);

    for (int layer = 0; layer < DEPTH; ++layer) {
        apply_layer_kernel<0, 0><<<1024, THREADS, ldsb, stream>>>(state, mats, layer, nullptr);
        if (layer < DEPTH - 1)
            apply_layer_kernel<1, 0><<<1024, THREADS, ldsb, stream>>>(state, mats, layer, nullptr);
        else
            apply_layer_kernel<1, 1><<<1024, THREADS, ldsb, stream>>>(state, mats, layer, out);
    }
}